// SecondaryCapsuleLayer_42872363549209
// MI455X (gfx1250) — compile-verified
//
#include <hip/hip_runtime.h>
#include <math.h>

// Problem constants (from reference setup_inputs)
#define BB 256
#define CC 10
#define RR 1152
#define II 8
#define OO 16
#define NITER 3

typedef __attribute__((ext_vector_type(2))) float v2f;
typedef __attribute__((ext_vector_type(8))) float v8f;

// LDS layout (floats), dynamic shared memory:
//  Ls     : 16*1152 = 18432   logits L[b][r]
//  red    : 256               16x16 partial-reduction scratch
//  rowmax : 16                softmax row max per b
//  rowinv : 16                1/sum(exp) per b
//  sred   : 8*32*8 = 2048     per-wave WMMA accumulator dump
//  sfin   : 256               s[b][o]
//  cf     : 16                squash coefficient per b
//  outt   : 256               out[b][o]
#define LDS_FLOATS (18432 + 256 + 16 + 16 + 2048 + 256 + 16 + 256)

__global__ __launch_bounds__(256, 1)
void capsule_route_fused(const float* __restrict__ x,   // [B, R, I]
                         const float* __restrict__ W,   // [C, R, I, O]
                         float* __restrict__ out)       // [B, C, O]
{
    extern __shared__ float smem[];
    float* Ls     = smem;             // 18432
    float* red    = Ls + 18432;       // 256
    float* rowmax = red + 256;        // 16
    float* rowinv = rowmax + 16;      // 16
    float* sred   = rowinv + 16;      // 2048
    float* sfin   = sred + 2048;      // 256
    float* cf     = sfin + 256;       // 16
    float* outt   = cf + 16;          // 256

    const int t     = threadIdx.x;          // 0..255, 8 wave32s
    const int lane  = t & 31;
    const int wave  = t >> 5;               // 0..7
    const int btile = blockIdx.x;           // 0..15  (16 batch rows each)
    const int c     = blockIdx.y;           // 0..9
    const int b0    = btile * 16;

    // thread -> (b, group) mapping for the scalar passes
    const int bt = t & 15;                  // batch row within tile
    const int g  = t >> 4;                  // 0..15, strides over r

    // WMMA lane mapping (ISA 7.12.2, 32-bit operands):
    //  A 16x4 : lanes 0-15 M=0..15 ; VGPR0 holds K=2h, VGPR1 holds K=2h+1 (h = lane>>4)
    //  B 4x16 : lanes 0-15 N=0..15 ; VGPR0 holds K=2h, VGPR1 holds K=2h+1
    //  C/D    : element (vgpr k, lane l) -> M = k + 8*(l>>4), N = l&15
    const int M = lane & 15;
    const int h = lane >> 4;
    const int N = lane & 15;

    // Zero logits
    for (int idx = t; idx < 16 * RR; idx += 256) Ls[idx] = 0.0f;
    __syncthreads();

    const float* xM = x + ((size_t)(b0 + M) * RR) * II;     // this lane's A row base
    const float* wc = W + (size_t)c * RR * (II * OO);       // W[c]
    const int r_lo = wave * (RR / 8);
    const int r_hi = r_lo + (RR / 8);

    for (int it = 0; it < NITER; ++it) {
        // ---- softmax statistics over r, per batch row b ----
        {
            float m = -INFINITY;
            for (int r = g; r < RR; r += 16) m = fmaxf(m, Ls[bt * RR + r]);
            red[g * 16 + bt] = m;
        }
        __syncthreads();
        if (t < 16) {
            float mm = red[t];
            for (int gg = 1; gg < 16; ++gg) mm = fmaxf(mm, red[gg * 16 + t]);
            rowmax[t] = mm;
        }
        __syncthreads();
        {
            float s = 0.0f;
            float rm = rowmax[bt];
            for (int r = g; r < RR; r += 16) s += __expf(Ls[bt * RR + r] - rm);
            red[g * 16 + bt] = s;
        }
        __syncthreads();
        if (t < 16) {
            float ssum = 0.0f;
            for (int gg = 0; gg < 16; ++gg) ssum += red[gg * 16 + t];
            rowinv[t] = 1.0f / ssum;
        }
        __syncthreads();

        // ---- s[b,o] = sum_r prob[b,r] * P[b,r,o] via chained fp32 WMMA ----
        // prob folded into A:  s = sum_r sum_i (prob[b,r]*x[b,r,i]) * W[c,r,i,o]
        v8f acc = {};
        {
            const float rm = rowmax[M];
            const float ri = rowinv[M];
            for (int r = r_lo; r < r_hi; ++r) {
                const float  p  = __expf(Ls[M * RR + r] - rm) * ri;
                const float* xr = xM + (size_t)r * II;
                const float* wr = wc + (size_t)r * (II * OO);
                __builtin_prefetch(wr + II * OO, 0, 1);   // next route's W line

                v2f a0, a1, f0, f1;
                a0.x = p * xr[2 * h];     a0.y = p * xr[2 * h + 1];
                a1.x = p * xr[4 + 2 * h]; a1.y = p * xr[5 + 2 * h];
                f0.x = wr[(2 * h) * OO + N];     f0.y = wr[(2 * h + 1) * OO + N];
                f1.x = wr[(4 + 2 * h) * OO + N]; f1.y = wr[(5 + 2 * h) * OO + N];

                acc = __builtin_amdgcn_wmma_f32_16x16x4_f32(
                          false, a0, false, f0, (short)0, acc, false, false);
                acc = __builtin_amdgcn_wmma_f32_16x16x4_f32(
                          false, a1, false, f1, (short)0, acc, false, false);
            }
        }
        // dump per-wave accumulators to LDS
        for (int k = 0; k < 8; ++k)
            sred[(wave * 32 + lane) * 8 + k] = acc[k];
        __syncthreads();

        // ---- cross-wave reduction:  sfin[b][o] = sum_w acc_w ----
        {
            const int l = t >> 3, k = t & 7;
            float v = 0.0f;
            for (int w = 0; w < 8; ++w) v += sred[(w * 32 + l) * 8 + k];
            const int b = k + 8 * (l >> 4);
            const int o = l & 15;
            sfin[b * OO + o] = v;
        }
        __syncthreads();

        // ---- squash coefficient per b ----
        if (t < 16) {
            float sj2 = 0.0f;
            for (int o = 0; o < OO; ++o) { float sv = sfin[t * OO + o]; sj2 += sv * sv; }
            cf[t] = sj2 / ((1.0f + sj2) * sqrtf(sj2));
        }
        __syncthreads();

        // ---- out[b][o] (or final store) ----
        {
            const int b = t >> 4, o = t & 15;
            const float ov = cf[b] * sfin[b * OO + o];
            if (it == NITER - 1)
                out[((size_t)(b0 + b) * CC + c) * OO + o] = ov;
            else
                outt[b * OO + o] = ov;
        }
        __syncthreads();

        // ---- L[b,r] += sum_o P[b,r,o]*out[b,o] = sum_i x[b,r,i]*(sum_o W[c,r,i,o]*out[b,o]) ----
        if (it != NITER - 1) {
            float ov[OO];
            for (int o = 0; o < OO; ++o) ov[o] = outt[bt * OO + o];
            const float* xb = x + ((size_t)(b0 + bt) * RR) * II;
            for (int r = g; r < RR; r += 16) {
                const float* wr = wc + (size_t)r * (II * OO);
                const float* xr = xb + (size_t)r * II;
                float dl = 0.0f;
                for (int i = 0; i < II; ++i) {
                    float wd = 0.0f;
                    for (int o = 0; o < OO; ++o) wd += wr[i * OO + o] * ov[o];
                    dl += xr[i] * wd;
                }
                Ls[bt * RR + r] += dl;
            }
            __syncthreads();
        }
    }
}

extern "C" void kernel_launch(void* const* d_in, const int* in_sizes, int n_in,
                              void* d_out, int out_size, void* d_ws, size_t ws_size,
                              hipStream_t stream) {
    (void)in_sizes; (void)n_in; (void)out_size; (void)d_ws; (void)ws_size;
    const float* x = (const float*)d_in[0];   // [256, 1152, 8]
    const float* W = (const float*)d_in[1];   // [10, 1152, 8, 16]
    float* out = (float*)d_out;               // [256, 10, 16]

    const size_t shmem = (size_t)LDS_FLOATS * sizeof(float);   // ~85 KB (CDNA5: 320 KB/WGP)
    hipFuncSetAttribute((const void*)capsule_route_fused,
                        hipFuncAttributeMaxDynamicSharedMemorySize, (int)shmem);

    dim3 grid(BB / 16, CC);   // 16 x 10 = 160 workgroups
    dim3 block(256);          // 8 wave32s
    capsule_route_fused<<<grid, block, shmem, stream>>>(x, W, out);
}